// GGCNlayer_SP_7129645711850
// MI455X (gfx1250) — compile-verified
//
#include <hip/hip_runtime.h>
#include <hip/hip_bf16.h>

typedef __attribute__((ext_vector_type(2))) float v2f;
typedef __attribute__((ext_vector_type(8))) float v8f;

#define D 64        // D_IN == D_OUT == 64
#define LSTR 66     // padded LDS row stride (dwords) -> conflict-free b64 loads

// ---------------------------------------------------------------------------
// Kernel 0: fold the tiny parameter nonlinearities once.
// consts = { c0, c1, c2, scale }, coeff = softmax(coeff_param),
// scale = softplus(scale_param).
// ---------------------------------------------------------------------------
__global__ void consts_kernel(const float* __restrict__ coeff_param,
                              const float* __restrict__ scale_param,
                              float* __restrict__ consts) {
    if (threadIdx.x != 0 || blockIdx.x != 0) return;
    float x0 = coeff_param[0], x1 = coeff_param[1], x2 = coeff_param[2];
    float m = fmaxf(x0, fmaxf(x1, x2));
    float e0 = expf(x0 - m), e1 = expf(x1 - m), e2 = expf(x2 - m);
    float inv = 1.0f / (e0 + e1 + e2);
    consts[0] = e0 * inv;
    consts[1] = e1 * inv;
    consts[2] = e2 * inv;
    float s = scale_param[0];
    consts[3] = (s > 20.0f) ? s : log1pf(expf(s));
}

// ---------------------------------------------------------------------------
// Kernel 1: Wh = h @ W + b via V_WMMA_F32_16X16X4_F32 (exact f32 path).
// One wave computes a 16x64 stripe: 4 n-tiles x 16 k-steps = 64 WMMA ops.
//
// W is staged TRANSPOSED in LDS (Wt[n][k], stride 66 dwords) so each B
// fragment {W[k][n], W[k+1][n]} is one contiguous, even-aligned ds_load_b64
// with conflict-free banking (bank = (2n+k) mod 64 across lanes 0-15).
//
// A-frag (16x4 f32): lanes 0-15 hold K={k0,k0+1}, lanes 16-31 hold K={k0+2,k0+3},
// row M = lane&15.  B-frag mirrors A over K, N = lane&15.
// C/D: VGPR v -> M = v + 8*(lane>>4), N = lane&15.
// ---------------------------------------------------------------------------
__global__ void __launch_bounds__(128)
gemm_wmma_kernel(const float* __restrict__ h, const float* __restrict__ W,
                 const float* __restrict__ bias, float* __restrict__ Wh, int N) {
    __shared__ float Ws[D * LSTR];

    int tid = threadIdx.x;
    // Stage W transposed: thread reads W[r][c..c+3], scatters into Wt[c][r].
    for (int i = tid * 4; i < D * D; i += 128 * 4) {
        int r = i >> 6;
        int c = i & 63;
        float4 w4 = *(const float4*)(W + i);
        Ws[(c + 0) * LSTR + r] = w4.x;
        Ws[(c + 1) * LSTR + r] = w4.y;
        Ws[(c + 2) * LSTR + r] = w4.z;
        Ws[(c + 3) * LSTR + r] = w4.w;
    }
    __syncthreads();

    int wave = tid >> 5;
    int lane = tid & 31;
    int half = lane >> 4;   // 0: K pair {k0,k0+1}; 1: K pair {k0+2,k0+3}
    int l16  = lane & 15;

    int m0 = (blockIdx.x * 4 + wave) * 16;
    if (m0 >= N) return;

    const float* hrow = h + (size_t)(m0 + l16) * D + 2 * half;
    const float* wt   = Ws + l16 * LSTR + 2 * half;   // + t*16*LSTR + 4*s

    v8f acc[4] = {};

#pragma unroll 4
    for (int s = 0; s < 16; ++s) {
        v2f a = *(const v2f*)(hrow + 4 * s);
#pragma unroll
        for (int t = 0; t < 4; ++t) {
            v2f bf = *(const v2f*)(wt + t * (16 * LSTR) + 4 * s);
            acc[t] = __builtin_amdgcn_wmma_f32_16x16x4_f32(
                false, a, false, bf, (short)0, acc[t], false, false);
        }
    }

#pragma unroll
    for (int t = 0; t < 4; ++t) {
        int n = t * 16 + l16;
        float bv = bias[n];
#pragma unroll
        for (int v = 0; v < 8; ++v) {
            Wh[(size_t)(m0 + half * 8 + v) * D + n] = acc[t][v] + bv;
        }
    }
}

// ---------------------------------------------------------------------------
// Kernel 2: per-node L2 norms of Wh rows.  16 lanes per node, float4/lane.
// ---------------------------------------------------------------------------
__global__ void __launch_bounds__(256)
norms_kernel(const float* __restrict__ Wh, float* __restrict__ norms, int N) {
    int gtid = blockIdx.x * 256 + threadIdx.x;
    int node = gtid >> 4;
    int l    = gtid & 15;
    if (node >= N) return;
    float4 v = *(const float4*)(Wh + (size_t)node * D + l * 4);
    float s = v.x * v.x + v.y * v.y + v.z * v.z + v.w * v.w;
    for (int m = 8; m >= 1; m >>= 1) s += __shfl_xor(s, m, 16);
    if (l == 0) norms[node] = sqrtf(s);
}

// ---------------------------------------------------------------------------
// Kernel 3: zero prop accumulator (harness does not re-zero between replays).
// ---------------------------------------------------------------------------
__global__ void __launch_bounds__(256)
zero_kernel(float4* __restrict__ p, int n4) {
    int i = blockIdx.x * 256 + threadIdx.x;
    if (i < n4) p[i] = make_float4(0.f, 0.f, 0.f, 0.f);
}

// ---------------------------------------------------------------------------
// Kernel 4: edge phase.  16 lanes cooperate on one edge:
//   dot = <Wh[row], Wh[col]> (float4/lane, shfl-xor width-16 reduction)
//   sim = dot / max(nr*nc, eps);  sc = softplus(dc0*deg + dc1)
//   w   = adj*sc * (sim>0 ? c0*sim : c1*sim)    (fuses pos+neg passes)
//   prop[row] += w * Wh[col]   (4x atomicAdd f32 per lane, L2-resident)
// ---------------------------------------------------------------------------
__global__ void __launch_bounds__(256)
edge_kernel(const int* __restrict__ ei, const float* __restrict__ adj_v,
            const float* __restrict__ deg_v, const float* __restrict__ Wh,
            const float* __restrict__ norms, const float* __restrict__ dcoef,
            const float* __restrict__ consts, float* __restrict__ prop, int E) {
    int tid = threadIdx.x;
    int l   = tid & 15;
    int eid = blockIdx.x * 16 + (tid >> 4);
    if (eid >= E) return;

    int row = ei[eid];
    int col = ei[E + eid];
    if (row == col) return;  // no_diag: contribution is exactly zero

    float4 r4 = *(const float4*)(Wh + (size_t)row * D + l * 4);
    float4 c4 = *(const float4*)(Wh + (size_t)col * D + l * 4);
    float dot = r4.x * c4.x + r4.y * c4.y + r4.z * c4.z + r4.w * c4.w;
    for (int m = 8; m >= 1; m >>= 1) dot += __shfl_xor(dot, m, 16);

    float nr = norms[row];
    float nc = norms[col];
    float sim = dot / fmaxf(nr * nc, 1e-8f);

    float x  = dcoef[0] * deg_v[eid] + dcoef[1];
    float sc = (x > 20.0f) ? x : log1pf(expf(x));

    float c0 = consts[0], c1 = consts[1];
    float w = adj_v[eid] * sc * (sim > 0.0f ? c0 * sim : c1 * sim);
    if (w == 0.0f) return;

    float* p = prop + (size_t)row * D + l * 4;
    atomicAdd(p + 0, w * c4.x);
    atomicAdd(p + 1, w * c4.y);
    atomicAdd(p + 2, w * c4.z);
    atomicAdd(p + 3, w * c4.w);
}

// ---------------------------------------------------------------------------
// Kernel 5: out = scale * (prop + c2 * Wh)
// ---------------------------------------------------------------------------
__global__ void __launch_bounds__(256)
final_kernel(const float4* __restrict__ prop, const float4* __restrict__ Wh,
             const float* __restrict__ consts, float4* __restrict__ out, int n4) {
    int i = blockIdx.x * 256 + threadIdx.x;
    if (i >= n4) return;
    float c2 = consts[2], scale = consts[3];
    float4 p = prop[i];
    float4 w = Wh[i];
    float4 o;
    o.x = scale * (p.x + c2 * w.x);
    o.y = scale * (p.y + c2 * w.y);
    o.z = scale * (p.z + c2 * w.z);
    o.w = scale * (p.w + c2 * w.w);
    out[i] = o;
}

// ---------------------------------------------------------------------------
extern "C" void kernel_launch(void* const* d_in, const int* in_sizes, int n_in,
                              void* d_out, int out_size, void* d_ws, size_t ws_size,
                              hipStream_t stream) {
    const float* h     = (const float*)d_in[0];
    const int*   ei    = (const int*)  d_in[1];
    const float* adj_v = (const float*)d_in[2];
    const float* deg_v = (const float*)d_in[3];
    const float* W     = (const float*)d_in[4];
    const float* bias  = (const float*)d_in[5];
    const float* dcoef = (const float*)d_in[6];
    const float* cpar  = (const float*)d_in[7];
    const float* spar  = (const float*)d_in[8];

    const int N = in_sizes[0] / D;   // 65536
    const int E = in_sizes[2];       // 1048576

    // Workspace layout: Wh[N*64] | prop[N*64] | norms[N] | consts[4]
    float* Wh     = (float*)d_ws;
    float* prop   = Wh + (size_t)N * D;
    float* norms  = prop + (size_t)N * D;
    float* consts = norms + N;

    consts_kernel<<<1, 1, 0, stream>>>(cpar, spar, consts);

    gemm_wmma_kernel<<<(N + 63) / 64, 128, 0, stream>>>(h, W, bias, Wh, N);

    norms_kernel<<<(N * 16 + 255) / 256, 256, 0, stream>>>(Wh, norms, N);

    int n4 = N * D / 4;
    zero_kernel<<<(n4 + 255) / 256, 256, 0, stream>>>((float4*)prop, n4);

    edge_kernel<<<(E + 15) / 16, 256, 0, stream>>>(ei, adj_v, deg_v, Wh, norms,
                                                   dcoef, consts, prop, E);

    final_kernel<<<(n4 + 255) / 256, 256, 0, stream>>>((const float4*)prop,
                                                       (const float4*)Wh, consts,
                                                       (float4*)d_out, n4);
}